// BatchTopK_65352222376484
// MI455X (gfx1250) — compile-verified
//
#include <hip/hip_runtime.h>
#include <stdint.h>

// ---------------------------------------------------------------------------
// BatchTopK via exact radix threshold selection.
// Survivors are positive floats -> uint bit-pattern order == float order.
// Pipeline: coarse 12b hist (full read, ascending) -> scan -> FUSED fine-hist
// + output (full read DESCENDING to reuse L2 tail from pass 1, NT stores,
// undecided bucket-B elems wave-aggregated into a compact list) -> scan ->
// list fixup -> overflow fallback (early-exit normally).
// Traffic ~= 2 reads + 1 write - ~190MB L2 hits ~= 580 MB ~= 25 us @23.3TB/s.
// ---------------------------------------------------------------------------

#define H2_BITS 19u                 // low radix bits
#define H1_SIZE 4096u               // 2^12 coarse buckets (bits [30:19])
#define H2_SIZE (1u << H2_BITS)     // 524288 fine buckets (bits [18:0])
#define REP 4u                      // LDS histogram replication
#define LIST_CAP (1u << 19)         // candidate list capacity (512K entries)

// d_ws layout in uint32 words
#define WS_H1   0u
#define WS_H2   (WS_H1 + H1_SIZE)
#define WS_CTRL (WS_H2 + H2_SIZE)   // [0]=B12 [1]=R [2]=Tbits [3]=listCount
#define WS_LIST (WS_CTRL + 16u)     // LIST_CAP * 2 words: (elemIdx, bits)
#define WS_ZERO_WORDS (WS_CTRL + 16u)

#define NB_STREAM 4096
#define NT_STREAM 256

#if defined(__has_builtin)
#if __has_builtin(__builtin_amdgcn_global_load_async_to_lds_b128) && \
    __has_builtin(__builtin_amdgcn_s_wait_asynccnt)
#define HAVE_ASYNC_LDS 1
#endif
#endif

typedef __attribute__((__vector_size__(4 * sizeof(int)))) int v4i;
typedef __attribute__((ext_vector_type(4))) float v4f;
#define AS_GLOBAL __attribute__((address_space(1)))
#define AS_LOCAL  __attribute__((address_space(3)))

// ---------------------------------------------------------------- zero ws --
__global__ __launch_bounds__(256) void k_zero(uint32_t* __restrict__ ws) {
    uint32_t i = blockIdx.x * blockDim.x + threadIdx.x;
    if (i < WS_ZERO_WORDS) ws[i] = 0u;
}

// ------------------------------------------------- pass 1: coarse 12b hist --
__global__ __launch_bounds__(NT_STREAM) void k_hist1(const float4* __restrict__ x4,
                                                     uint32_t* __restrict__ ws,
                                                     uint32_t n4) {
    __shared__ uint32_t lh[REP * H1_SIZE];           // 64 KB LDS
    for (uint32_t i = threadIdx.x; i < REP * H1_SIZE; i += blockDim.x) lh[i] = 0u;
    __syncthreads();

    const uint32_t rep_base = ((threadIdx.x >> 5) & (REP - 1u)) * H1_SIZE;
    const uint32_t stride = gridDim.x * blockDim.x;
    for (uint32_t i = blockIdx.x * blockDim.x + threadIdx.x; i < n4; i += stride) {
        if (i + stride < n4) __builtin_prefetch(&x4[i + stride], 0, 0);
        float4 v = x4[i];
        float c[4] = {v.x, v.y, v.z, v.w};
#pragma unroll
        for (int j = 0; j < 4; ++j) {
            if (c[j] > 0.0f) {
                uint32_t b = __float_as_uint(c[j]) >> H2_BITS;   // sign=0 -> <4096
                atomicAdd(&lh[rep_base + b], 1u);
            }
        }
    }
    __syncthreads();

    for (uint32_t b = threadIdx.x; b < H1_SIZE; b += blockDim.x) {
        uint32_t s = 0u;
#pragma unroll
        for (uint32_t r = 0; r < REP; ++r) s += lh[r * H1_SIZE + b];
        if (s) atomicAdd(&ws[WS_H1 + b], s);
    }
}

// --------------------------------- scan 1: find coarse bucket B, residual R --
__global__ __launch_bounds__(256) void k_scan1(const int* __restrict__ kp,
                                               uint32_t nrows,
                                               uint32_t* __restrict__ ws) {
    __shared__ uint32_t cs[256];
    const uint32_t CH = H1_SIZE / 256u;              // 16 buckets per chunk
    const uint32_t c = threadIdx.x;                  // chunk 0 = topmost buckets
    uint32_t s = 0u;
    const uint32_t base = H1_SIZE - CH * (c + 1u);
    for (uint32_t j = 0; j < CH; ++j) s += ws[WS_H1 + base + j];
    cs[c] = s;
    __syncthreads();

    if (threadIdx.x == 0) {
        const uint64_t ktotal = (uint64_t)kp[0] * (uint64_t)nrows;
        uint64_t cum = 0;
        int tc = -1;
        for (int cc = 0; cc < 256; ++cc) {
            if (cum + cs[cc] >= ktotal) { tc = cc; break; }
            cum += cs[cc];
        }
        uint32_t B = 0u, R = 0xFFFFFFFFu;            // default: keep all positives
        if (tc >= 0) {
            const int lo = (int)(H1_SIZE - CH * (uint32_t)(tc + 1));
            for (int b = (int)(H1_SIZE - CH * (uint32_t)tc) - 1; b >= lo; --b) {
                uint32_t h = ws[WS_H1 + (uint32_t)b];
                if (cum + h >= ktotal) {             // cum == count strictly above b
                    B = (uint32_t)b;
                    R = (uint32_t)(ktotal - cum);    // >= 1 needed from bucket B
                    break;
                }
                cum += h;
            }
        }
        ws[WS_CTRL + 0] = B;
        ws[WS_CTRL + 1] = R;
    }
}

// ------------- fused pass 2 element: decide / record (wave-aggregated) ------
__device__ __forceinline__ float fused_elem(float v, uint32_t elem_idx, uint32_t B,
                                            uint32_t* __restrict__ ws) {
    float r = 0.0f;
    bool cand = false;
    uint32_t u = 0u;
    if (v > 0.0f) {
        u = __float_as_uint(v);
        const uint32_t cb = u >> H2_BITS;
        if (cb > B) r = v;                           // definitely kept
        cand = (cb == B);                            // undecided
    }
    if (cand) atomicAdd(&ws[WS_H2 + (u & (H2_SIZE - 1u))], 1u);
    // wave32-aggregated append to candidate list (one atomic per wave)
    const uint32_t m32 = (uint32_t)__ballot((int)cand);
    if (m32 != 0u) {
        const uint32_t lane = __lane_id();
        const uint32_t prefix = __popc(m32 & ((1u << lane) - 1u));
        const int leader = __ffs(m32) - 1;
        uint32_t base = 0u;
        if ((int)lane == leader) base = atomicAdd(&ws[WS_CTRL + 3], __popc(m32));
        base = (uint32_t)__shfl((int)base, leader, 32);
        if (cand) {
            const uint32_t pos = base + prefix;
            if (pos < LIST_CAP) {
                ws[WS_LIST + 2u * pos + 0u] = elem_idx;
                ws[WS_LIST + 2u * pos + 1u] = u;
            }
        }
    }
    return r;
}

// ----- fused pass 2: DESCENDING slices (L2 tail reuse), NT output stores ----
__global__ __launch_bounds__(NT_STREAM) void k_fused(const float4* __restrict__ x4,
                                                     float4* __restrict__ out4,
                                                     uint32_t* __restrict__ ws,
                                                     uint32_t n4) {
    const uint32_t B = ws[WS_CTRL + 0];
    const uint32_t stride = gridDim.x * blockDim.x;
    const uint32_t base0 = blockIdx.x * blockDim.x + threadIdx.x;
    const uint32_t M = (n4 + stride - 1u) / stride;  // slice count

#if defined(HAVE_ASYNC_LDS)
    __shared__ float stage[2][NT_STREAM * 4];        // 8 KB, double buffered
    int buf = 0;
    {
        const uint32_t j0 = (M - 1u) * stride + base0;   // top slice first
        if (j0 < n4) {
            __builtin_amdgcn_global_load_async_to_lds_b128(
                (AS_GLOBAL v4i*)(x4 + j0),
                (AS_LOCAL  v4i*)&stage[buf][threadIdx.x * 4], 0, 0);
        }
    }
    for (uint32_t m = 0; m < M; ++m) {
        const uint32_t j = (M - 1u - m) * stride + base0;
        if (m + 1u < M) {
            const uint32_t jn = (M - 2u - m) * stride + base0;  // always < n4
            __builtin_amdgcn_global_load_async_to_lds_b128(
                (AS_GLOBAL v4i*)(x4 + jn),
                (AS_LOCAL  v4i*)&stage[buf ^ 1][threadIdx.x * 4], 0, 0);
            __builtin_amdgcn_s_wait_asynccnt(1);     // current buffer landed
        } else {
            __builtin_amdgcn_s_wait_asynccnt(0);
        }
        if (j < n4) {
            float c[4];
            c[0] = stage[buf][threadIdx.x * 4 + 0];
            c[1] = stage[buf][threadIdx.x * 4 + 1];
            c[2] = stage[buf][threadIdx.x * 4 + 2];
            c[3] = stage[buf][threadIdx.x * 4 + 3];
            v4f r;
            r.x = fused_elem(c[0], j * 4u + 0u, B, ws);
            r.y = fused_elem(c[1], j * 4u + 1u, B, ws);
            r.z = fused_elem(c[2], j * 4u + 2u, B, ws);
            r.w = fused_elem(c[3], j * 4u + 3u, B, ws);
            __builtin_nontemporal_store(r, (v4f*)(out4 + j));
        }
        buf ^= 1;
    }
#else
    for (uint32_t m = 0; m < M; ++m) {
        const uint32_t j = (M - 1u - m) * stride + base0;
        if (j < n4) {
            float4 v = x4[j];
            v4f r;
            r.x = fused_elem(v.x, j * 4u + 0u, B, ws);
            r.y = fused_elem(v.y, j * 4u + 1u, B, ws);
            r.z = fused_elem(v.z, j * 4u + 2u, B, ws);
            r.w = fused_elem(v.w, j * 4u + 3u, B, ws);
            __builtin_nontemporal_store(r, (v4f*)(out4 + j));
        }
    }
#endif
}

// ----------------------------------- scan 2: exact 32-bit threshold T bits --
__global__ __launch_bounds__(1024) void k_scan2(uint32_t* __restrict__ ws) {
    __shared__ uint32_t cs[1024];
    __shared__ uint32_t cv[512];
    __shared__ int s_tc;
    const uint32_t CH = H2_SIZE / 1024u;             // 512 buckets per chunk
    const uint32_t c = threadIdx.x;
    {
        uint32_t s = 0u;
        const uint32_t base = H2_SIZE - CH * (c + 1u);
        for (uint32_t j = 0; j < CH; ++j) s += ws[WS_H2 + base + j];
        cs[c] = s;
    }
    __syncthreads();

    uint64_t cum = 0;                                // live only in thread 0
    if (threadIdx.x == 0) {
        const uint32_t R = ws[WS_CTRL + 1];
        int tc = -1;
        for (int cc = 0; cc < 1024; ++cc) {
            if (cum + cs[cc] >= (uint64_t)R) { tc = cc; break; }
            cum += cs[cc];
        }
        s_tc = tc;
    }
    __syncthreads();

    const int tc = s_tc;
    if (tc >= 0) {
        const uint32_t cbase = H2_SIZE - CH * (uint32_t)(tc + 1);
        if (threadIdx.x < CH) cv[threadIdx.x] = ws[WS_H2 + cbase + threadIdx.x];
        __syncthreads();
        if (threadIdx.x == 0) {
            const uint32_t R = ws[WS_CTRL + 1];
            const uint32_t B = ws[WS_CTRL + 0];
            uint32_t T = 0u;
            for (int l = (int)CH - 1; l >= 0; --l) {
                uint32_t h = cv[l];
                if (cum + h >= (uint64_t)R) {
                    T = (B << H2_BITS) | (cbase + (uint32_t)l);
                    break;
                }
                cum += h;
            }
            ws[WS_CTRL + 2] = T;
        }
    } else {
        __syncthreads();
        if (threadIdx.x == 0) ws[WS_CTRL + 2] = 0u;  // keep all positives
    }
}

// -------------------- fixup: rewrite recorded bucket-B survivors -----------
__global__ __launch_bounds__(256) void k_fixup(float* __restrict__ out,
                                               const uint32_t* __restrict__ ws) {
    const uint32_t T = ws[WS_CTRL + 2];
    uint32_t cnt = ws[WS_CTRL + 3];
    if (cnt > LIST_CAP) cnt = LIST_CAP;
    const uint32_t i = blockIdx.x * blockDim.x + threadIdx.x;
    if (i < cnt) {
        const uint32_t idx = ws[WS_LIST + 2u * i + 0u];
        const uint32_t u   = ws[WS_LIST + 2u * i + 1u];
        if (u >= T) out[idx] = __uint_as_float(u);
    }
}

// ------------- overflow fallback: full corrective pass (normally exits) ----
__global__ __launch_bounds__(NT_STREAM) void k_fallback(const float4* __restrict__ x4,
                                                        float* __restrict__ out,
                                                        const uint32_t* __restrict__ ws,
                                                        uint32_t n4) {
    if (ws[WS_CTRL + 3] <= LIST_CAP) return;         // common case: nothing to do
    const uint32_t B = ws[WS_CTRL + 0];
    const uint32_t T = ws[WS_CTRL + 2];
    const uint32_t stride = gridDim.x * blockDim.x;
    for (uint32_t i = blockIdx.x * blockDim.x + threadIdx.x; i < n4; i += stride) {
        float4 v = x4[i];
        float c[4] = {v.x, v.y, v.z, v.w};
#pragma unroll
        for (int j = 0; j < 4; ++j) {
            if (c[j] > 0.0f) {
                const uint32_t u = __float_as_uint(c[j]);
                if ((u >> H2_BITS) == B && u >= T) out[i * 4u + (uint32_t)j] = c[j];
            }
        }
    }
}

// ---------------------------------------------------------------------------
extern "C" void kernel_launch(void* const* d_in, const int* in_sizes, int n_in,
                              void* d_out, int out_size, void* d_ws, size_t ws_size,
                              hipStream_t stream) {
    const float4* x4 = (const float4*)d_in[0];
    const int* kp = (const int*)d_in[1];           // k lives in device memory
    const uint32_t n  = (uint32_t)in_sizes[0];     // 1024 * 65536
    const uint32_t n4 = n >> 2;
    const uint32_t nrows = n >> 16;                // d_sae = 65536 columns
    uint32_t* ws = (uint32_t*)d_ws;
    float4* out4 = (float4*)d_out;
    float* out = (float*)d_out;

    k_zero    <<<(WS_ZERO_WORDS + 255) / 256, 256, 0, stream>>>(ws);
    k_hist1   <<<NB_STREAM, NT_STREAM, 0, stream>>>(x4, ws, n4);
    k_scan1   <<<1, 256, 0, stream>>>(kp, nrows, ws);
    k_fused   <<<NB_STREAM, NT_STREAM, 0, stream>>>(x4, out4, ws, n4);
    k_scan2   <<<1, 1024, 0, stream>>>(ws);
    k_fixup   <<<LIST_CAP / 256, 256, 0, stream>>>(out, ws);
    k_fallback<<<NB_STREAM, NT_STREAM, 0, stream>>>(x4, out, ws, n4);
}